// GCN_84670985273813
// MI455X (gfx1250) — compile-verified
//
#include <hip/hip_runtime.h>
#include <stdint.h>

typedef __attribute__((ext_vector_type(2))) float v2f;
typedef __attribute__((ext_vector_type(8))) float v8f;

#define N_NODES 100000
#define N_EDGES 3200000
#define F_IN    256
#define F_HID   64
#define F_OUT   40
#define F_OUT_PAD 48

// ---------------- utility kernels ----------------

__global__ void k_fill_f32(float* __restrict__ p, float v, int n) {
  int i = blockIdx.x * blockDim.x + threadIdx.x;
  if (i < n) p[i] = v;
}

__global__ void k_accum_deg(const int* __restrict__ dst, float* __restrict__ deg) {
  int e = blockIdx.x * blockDim.x + threadIdx.x;
  if (e < N_EDGES) atomicAdd(deg + dst[e], 1.0f);
}

__global__ void k_rsqrt_inplace(float* __restrict__ deg) {
  int i = blockIdx.x * blockDim.x + threadIdx.x;
  if (i < N_NODES) deg[i] = rsqrtf(deg[i]);   // deg >= 1 always (self loop)
}

// ---------------- GEMM1: h1[N,64] = x[N,256] @ W1[256,64] via f32 WMMA ----------------
// block = 128 threads (4 waves); block tile = 16 rows x 64 cols; wave tile = 16x16.
__global__ __launch_bounds__(128) void k_gemm1_wmma(const float* __restrict__ x,
                                                    const float* __restrict__ W1,
                                                    float* __restrict__ h1) {
  __shared__ float WT[F_HID][F_IN];           // W1 transposed: 64*256*4 = 64 KB
  const int tid = threadIdx.x;
  for (int i = tid; i < F_IN * F_HID; i += 128) {
    int k = i >> 6;                            // W1 is [256,64] row-major
    int n = i & 63;
    WT[n][k] = W1[i];
  }
  __syncthreads();

  const int wave = tid >> 5;
  const int lane = tid & 31;
  const int ml   = lane & 15;                  // M (A) / N (B) within tile
  const int kh   = (lane >> 4) << 1;           // K half: 0 or 2
  const int n0   = wave * 16;
  const int m0   = blockIdx.x * 16;            // 100000 / 16 = 6250 exact

  const float* arow = x + (size_t)(m0 + ml) * F_IN + kh;
  const float* brow = &WT[n0 + ml][kh];

  v8f acc = {};
#pragma unroll
  for (int k = 0; k < F_IN; k += 4) {
    v2f a = *(const v2f*)(arow + k);           // x row is contiguous, 8B aligned
    v2f b = *(const v2f*)(brow + k);           // LDS, contiguous in K (transposed)
    acc = __builtin_amdgcn_wmma_f32_16x16x4_f32(false, a, false, b,
                                                (short)0, acc, false, false);
  }

  // C/D layout: VGPR r -> row r (lanes 0-15) / row r+8 (lanes 16-31), col = lane&15
  float* orow = h1 + (size_t)(m0 + ((lane >> 4) << 3)) * F_HID + n0 + ml;
#pragma unroll
  for (int r = 0; r < 8; ++r)
    orow[(size_t)r * F_HID] = acc[r];
}

// ---------------- edge aggregation layer 1: agg1[dst] += dinv[s]*dinv[d]*h1[src] ----
__global__ void k_agg1(const int* __restrict__ src, const int* __restrict__ dst,
                       const float* __restrict__ dinv, const float* __restrict__ h1,
                       float* __restrict__ agg) {
  long long idx = (long long)blockIdx.x * blockDim.x + threadIdx.x;
  if (idx >= (long long)N_EDGES * F_HID) return;
  int e = (int)(idx >> 6);
  int f = (int)(idx & 63);
  int s = src[e], d = dst[e];
  float nrm = dinv[s] * dinv[d];
  atomicAdd(&agg[(size_t)d * F_HID + f], nrm * h1[(size_t)s * F_HID + f]);
}

// ---------------- self-loop + bias + ReLU in place on agg1 ----------------
__global__ void k_finalize1(const float* __restrict__ dinv, const float* __restrict__ h1,
                            const float* __restrict__ b1, float* __restrict__ agg) {
  int i = blockIdx.x * blockDim.x + threadIdx.x;
  if (i >= N_NODES * F_HID) return;
  int node = i >> 6, f = i & 63;
  float sl = dinv[node]; sl *= sl;             // self-loop norm = dinv^2
  float v = agg[i] + sl * h1[i] + b1[f];
  agg[i] = v > 0.0f ? v : 0.0f;
}

// ---------------- GEMM2: h2[N,40] = relu1[N,64] @ W2[64,40] via f32 WMMA ------------
// block = 96 threads (3 waves cover N padded to 48); 16 rows per block.
__global__ __launch_bounds__(96) void k_gemm2_wmma(const float* __restrict__ hin,
                                                   const float* __restrict__ W2,
                                                   float* __restrict__ h2) {
  __shared__ float WT[F_OUT_PAD][F_HID];       // 48*64*4 = 12 KB, zero padded
  const int tid = threadIdx.x;
  for (int i = tid; i < F_OUT_PAD * F_HID; i += 96) {
    int n = i >> 6;
    int k = i & 63;
    WT[n][k] = (n < F_OUT) ? W2[k * F_OUT + n] : 0.0f;
  }
  __syncthreads();

  const int wave = tid >> 5;
  const int lane = tid & 31;
  const int ml   = lane & 15;
  const int kh   = (lane >> 4) << 1;
  const int n0   = wave * 16;
  const int m0   = blockIdx.x * 16;

  const float* arow = hin + (size_t)(m0 + ml) * F_HID + kh;
  const float* brow = &WT[n0 + ml][kh];

  v8f acc = {};
#pragma unroll
  for (int k = 0; k < F_HID; k += 4) {
    v2f a = *(const v2f*)(arow + k);
    v2f b = *(const v2f*)(brow + k);
    acc = __builtin_amdgcn_wmma_f32_16x16x4_f32(false, a, false, b,
                                                (short)0, acc, false, false);
  }

  int col = n0 + ml;
  if (col < F_OUT) {                           // divergence only after all WMMAs
    float* orow = h2 + (size_t)(m0 + ((lane >> 4) << 3)) * F_OUT + col;
#pragma unroll
    for (int r = 0; r < 8; ++r)
      orow[(size_t)r * F_OUT] = acc[r];
  }
}

// ---------------- edge aggregation layer 2 into d_out ----------------
__global__ void k_agg2(const int* __restrict__ src, const int* __restrict__ dst,
                       const float* __restrict__ dinv, const float* __restrict__ h2,
                       float* __restrict__ agg) {
  unsigned idx = blockIdx.x * blockDim.x + threadIdx.x;
  if (idx >= (unsigned)N_EDGES * F_OUT) return;
  unsigned e = idx / (unsigned)F_OUT;
  unsigned f = idx - e * (unsigned)F_OUT;
  int s = src[e], d = dst[e];
  float nrm = dinv[s] * dinv[d];
  atomicAdd(&agg[(size_t)d * F_OUT + f], nrm * h2[(size_t)s * F_OUT + f]);
}

// ---------------- self-loop + bias + log_softmax in place on d_out ----------------
__global__ void k_finalize2(const float* __restrict__ dinv, const float* __restrict__ h2,
                            const float* __restrict__ b2, float* __restrict__ out) {
  int i = blockIdx.x * blockDim.x + threadIdx.x;
  if (i >= N_NODES) return;
  float sl = dinv[i]; sl *= sl;
  float z[F_OUT];
  float mx = -3.402823466e38f;
#pragma unroll
  for (int f = 0; f < F_OUT; ++f) {
    float v = out[(size_t)i * F_OUT + f] + sl * h2[(size_t)i * F_OUT + f] + b2[f];
    z[f] = v;
    mx = fmaxf(mx, v);
  }
  float s = 0.0f;
#pragma unroll
  for (int f = 0; f < F_OUT; ++f) s += __expf(z[f] - mx);
  float lse = mx + __logf(s);
#pragma unroll
  for (int f = 0; f < F_OUT; ++f) out[(size_t)i * F_OUT + f] = z[f] - lse;
}

// ---------------- host side ----------------

extern "C" void kernel_launch(void* const* d_in, const int* in_sizes, int n_in,
                              void* d_out, int out_size, void* d_ws, size_t ws_size,
                              hipStream_t stream) {
  const float* x  = (const float*)d_in[0];
  const int*   ei = (const int*)d_in[1];      // [2, N_EDGES]
  const float* W1 = (const float*)d_in[2];
  const float* b1 = (const float*)d_in[3];
  const float* W2 = (const float*)d_in[4];
  const float* b2 = (const float*)d_in[5];
  const int* src = ei;
  const int* dst = ei + N_EDGES;
  float* out = (float*)d_out;

  // workspace layout (all 256B aligned):
  //   dinv [100000]       : 400128 B
  //   h1   [100000 x 64]  : 25,600,000 B   (reused as h2 [100000 x 40] later)
  //   agg1 [100000 x 64]  : 25,600,000 B
  char* ws = (char*)d_ws;
  float* dinv = (float*)ws;
  float* h1   = (float*)(ws + 400128);
  float* agg1 = (float*)(ws + 400128 + 25600000);
  float* h2   = h1;                            // h1 dead after finalize1

  const int T = 256;

  // degrees -> dinv (self loop counts as 1)
  k_fill_f32<<<(N_NODES + T - 1) / T, T, 0, stream>>>(dinv, 1.0f, N_NODES);
  k_accum_deg<<<(N_EDGES + T - 1) / T, T, 0, stream>>>(dst, dinv);
  k_rsqrt_inplace<<<(N_NODES + T - 1) / T, T, 0, stream>>>(dinv);

  // zero accumulators (must happen every call: harness does not re-poison)
  k_fill_f32<<<(N_NODES * F_HID + T - 1) / T, T, 0, stream>>>(agg1, 0.0f, N_NODES * F_HID);
  k_fill_f32<<<(N_NODES * F_OUT + T - 1) / T, T, 0, stream>>>(out, 0.0f, N_NODES * F_OUT);

  // layer 1
  k_gemm1_wmma<<<N_NODES / 16, 128, 0, stream>>>(x, W1, h1);
  long long w1 = (long long)N_EDGES * F_HID;
  k_agg1<<<(unsigned)((w1 + T - 1) / T), T, 0, stream>>>(src, dst, dinv, h1, agg1);
  k_finalize1<<<(N_NODES * F_HID + T - 1) / T, T, 0, stream>>>(dinv, h1, b1, agg1);

  // layer 2
  k_gemm2_wmma<<<N_NODES / 16, 96, 0, stream>>>(agg1, W2, h2);
  unsigned w2 = (unsigned)N_EDGES * F_OUT;
  k_agg2<<<(w2 + T - 1) / T, T, 0, stream>>>(src, dst, dinv, h2, out);
  k_finalize2<<<(N_NODES + T - 1) / T, T, 0, stream>>>(dinv, h2, b2, out);
}